// GAE_43662637531914
// MI455X (gfx1250) — compile-verified
//
#include <hip/hip_runtime.h>
#include <cstdint>

// ---------------------------------------------------------------------------
// GAE forward for MI455X (gfx1250, wave32, WMMA).
//   x:(10000,128) edge_index:(2,640000) int64  W1:(128,32) b1:(32)
//   W2:(32,16) b2:(16)   ->  logits:(10000,10000) fp32
// Dominant cost: z @ z^T (3.2 GFLOP, 400MB out) -> V_WMMA_F32_16X16X4_F32
// tiles + non-temporal streaming stores. 10000 = 625*16 -> no padding.
// ---------------------------------------------------------------------------

typedef __attribute__((ext_vector_type(2))) float v2f;
typedef __attribute__((ext_vector_type(8))) float v8f;

#define N_NODES 10000
#define IN_DIM  128
#define H1_DIM  32
#define H2_DIM  16
#define N_EDGES 640000
#define MTILES  625   // 10000 / 16

// ---------------- degree / normalization --------------------------------
__global__ void k_deg_init(float* __restrict__ deg, int n) {
    int i = blockIdx.x * blockDim.x + threadIdx.x;
    if (i < n) deg[i] = 1.0f;                      // self-loop contribution
}

__global__ void k_deg_count(const int64_t* __restrict__ col,
                            float* __restrict__ deg, int e) {
    int i = blockIdx.x * blockDim.x + threadIdx.x;
    if (i < e) atomicAdd(&deg[(int)col[i]], 1.0f);
}

__global__ void k_dinv(const float* __restrict__ deg,
                       float* __restrict__ dinv, int n) {
    int i = blockIdx.x * blockDim.x + threadIdx.x;
    if (i < n) {
        float d = deg[i];
        dinv[i] = d > 0.0f ? __frsqrt_rn(d) : 0.0f;
    }
}

// ---------------- fp32 WMMA GEMM: C[M,N] = A[M,K] * B[K,N] --------------
// One wave per 16x16 output tile; K stepped by 4 via V_WMMA_F32_16X16X4_F32.
// A 16x4 layout: lanes 0-15 hold M=lane; VGPR0 = K=(0|2), VGPR1 = K=(1|3)
//                (lane half selects +2). B 4x16 mirrors across lanes.
// C/D 16x16: VGPR j -> M = j + 8*half, N = lane&15.
__global__ void __launch_bounds__(32)
k_gemm16_wmma(const float* __restrict__ A, const float* __restrict__ B,
              float* __restrict__ C, int N, int K) {
    const int lane = threadIdx.x;        // 0..31, EXEC all ones (WMMA req.)
    const int half = lane >> 4;          // 0 or 1
    const int l    = lane & 15;
    const int m0 = blockIdx.x * 16;
    const int n0 = blockIdx.y * 16;

    v8f acc = {};
    const float* arow = A + (size_t)(m0 + l) * K;
    for (int k = 0; k < K; k += 4) {
        v2f a, b;
        a.x = arow[k + 2 * half];
        a.y = arow[k + 2 * half + 1];
        b.x = B[(size_t)(k + 2 * half) * N + n0 + l];
        b.y = B[(size_t)(k + 2 * half + 1) * N + n0 + l];
        acc = __builtin_amdgcn_wmma_f32_16x16x4_f32(
            false, a, false, b, (short)0, acc, false, false);
    }
#pragma unroll
    for (int j = 0; j < 8; ++j)
        C[(size_t)(m0 + j + 8 * half) * N + n0 + l] = acc[j];
}

// ---------------- GCN aggregation ---------------------------------------
// out[i][f] = hlin[i][f] * dinv[i]^2   (self-loop term; also zero-inits out)
template <int F>
__global__ void k_selfloop(const float* __restrict__ hlin,
                           const float* __restrict__ dinv,
                           float* __restrict__ out, int n) {
    int t = blockIdx.x * blockDim.x + threadIdx.x;
    int i = t / F, f = t % F;
    if (i < n) {
        float di = dinv[i];
        out[(size_t)i * F + f] = hlin[(size_t)i * F + f] * di * di;
    }
}

// out[col][f] += hlin[row][f] * dinv[row] * dinv[col]   (1.28MB acc -> L2)
template <int F>
__global__ void k_scatter(const int64_t* __restrict__ row,
                          const int64_t* __restrict__ col,
                          const float* __restrict__ hlin,
                          const float* __restrict__ dinv,
                          float* __restrict__ out, int e) {
    long t = (long)blockIdx.x * blockDim.x + threadIdx.x;
    long ei = t / F;
    int  f  = (int)(t % F);
    if (ei < e) {
        int r = (int)row[ei], c = (int)col[ei];
        float nm = dinv[r] * dinv[c];
        atomicAdd(&out[(size_t)c * F + f], hlin[(size_t)r * F + f] * nm);
    }
}

template <int F, bool RELU>
__global__ void k_bias_act(float* __restrict__ h, const float* __restrict__ b,
                           int n) {
    int t = blockIdx.x * blockDim.x + threadIdx.x;
    int i = t / F, f = t % F;
    if (i < n) {
        float v = h[(size_t)i * F + f] + b[f];
        h[(size_t)i * F + f] = RELU ? fmaxf(v, 0.0f) : v;
    }
}

// ---------------- decode: logits = Z * Z^T  (K=16, store-BW bound) ------
// One wave per 16x16 logits tile; 4 WMMA steps; non-temporal stores so the
// 400MB output streams past L2 instead of evicting z / the edge accumulators.
__global__ void __launch_bounds__(32)
k_decode_wmma(const float* __restrict__ Z, float* __restrict__ L) {
    const int lane = threadIdx.x;
    const int half = lane >> 4;
    const int l    = lane & 15;
    const int m0 = blockIdx.x * 16;
    const int n0 = blockIdx.y * 16;

    v8f acc = {};
    const float* arow = Z + (size_t)(m0 + l) * H2_DIM;
    const float* brow = Z + (size_t)(n0 + l) * H2_DIM;   // B[k][n] = Z[n][k]
#pragma unroll
    for (int k = 0; k < H2_DIM; k += 4) {
        v2f a, b;
        a.x = arow[k + 2 * half];
        a.y = arow[k + 2 * half + 1];
        b.x = brow[k + 2 * half];
        b.y = brow[k + 2 * half + 1];
        acc = __builtin_amdgcn_wmma_f32_16x16x4_f32(
            false, a, false, b, (short)0, acc, false, false);
    }
#pragma unroll
    for (int j = 0; j < 8; ++j) {
        float* p = L + (size_t)(m0 + j + 8 * half) * N_NODES + n0 + l;
        __builtin_nontemporal_store(acc[j], p);
    }
}

// ---------------------------------------------------------------------------
extern "C" void kernel_launch(void* const* d_in, const int* in_sizes, int n_in,
                              void* d_out, int out_size, void* d_ws, size_t ws_size,
                              hipStream_t stream) {
    const float*   x    = (const float*)d_in[0];
    const int64_t* ei   = (const int64_t*)d_in[1];     // (2, E) int64
    const float*   W1   = (const float*)d_in[2];
    const float*   b1   = (const float*)d_in[3];
    const float*   W2   = (const float*)d_in[4];
    const float*   b2   = (const float*)d_in[5];
    float*         out  = (float*)d_out;               // (N, N) logits

    const int64_t* row = ei;
    const int64_t* col = ei + N_EDGES;

    // workspace carve-up (~4 MB)
    float* ws    = (float*)d_ws;
    float* deg   = ws;                       ws += 10016;
    float* dinv  = ws;                       ws += 10016;
    float* h1lin = ws;                       ws += (size_t)N_NODES * H1_DIM;
    float* h1    = ws;                       ws += (size_t)N_NODES * H1_DIM;
    float* h2lin = ws;                       ws += (size_t)N_NODES * H2_DIM;
    float* z     = ws;

    const int TB = 256;

    // 1) degrees + symmetric norm
    k_deg_init<<<(N_NODES + TB - 1) / TB, TB, 0, stream>>>(deg, N_NODES);
    k_deg_count<<<(N_EDGES + TB - 1) / TB, TB, 0, stream>>>(col, deg, N_EDGES);
    k_dinv<<<(N_NODES + TB - 1) / TB, TB, 0, stream>>>(deg, dinv, N_NODES);

    // 2) layer 1: h1lin = x @ W1  (WMMA f32, 625x2 tiles, K=128)
    k_gemm16_wmma<<<dim3(MTILES, H1_DIM / 16), 32, 0, stream>>>(
        x, W1, h1lin, H1_DIM, IN_DIM);
    k_selfloop<H1_DIM><<<(N_NODES * H1_DIM + TB - 1) / TB, TB, 0, stream>>>(
        h1lin, dinv, h1, N_NODES);
    {
        long tot = (long)N_EDGES * H1_DIM;
        k_scatter<H1_DIM><<<(unsigned)((tot + TB - 1) / TB), TB, 0, stream>>>(
            row, col, h1lin, dinv, h1, N_EDGES);
    }
    k_bias_act<H1_DIM, true><<<(N_NODES * H1_DIM + TB - 1) / TB, TB, 0, stream>>>(
        h1, b1, N_NODES);

    // 3) layer 2: h2lin = h1 @ W2  (WMMA f32, 625x1 tiles, K=32)
    k_gemm16_wmma<<<dim3(MTILES, H2_DIM / 16), 32, 0, stream>>>(
        h1, W2, h2lin, H2_DIM, H1_DIM);
    k_selfloop<H2_DIM><<<(N_NODES * H2_DIM + TB - 1) / TB, TB, 0, stream>>>(
        h2lin, dinv, z, N_NODES);
    {
        long tot = (long)N_EDGES * H2_DIM;
        k_scatter<H2_DIM><<<(unsigned)((tot + TB - 1) / TB), TB, 0, stream>>>(
            row, col, h2lin, dinv, z, N_EDGES);
    }
    k_bias_act<H2_DIM, false><<<(N_NODES * H2_DIM + TB - 1) / TB, TB, 0, stream>>>(
        z, b2, N_NODES);

    // 4) decode: logits = z @ z^T  (store-bandwidth bound, NT stores)
    k_decode_wmma<<<dim3(MTILES, MTILES), 32, 0, stream>>>(z, out);
}